// GraphSAGEModel_83511344103498
// MI455X (gfx1250) — compile-verified
//
#include <hip/hip_runtime.h>

typedef float v2f __attribute__((ext_vector_type(2)));
typedef float v8f __attribute__((ext_vector_type(8)));

#define D_FEAT 128            // K dimension for both layers
#define LDS_STRIDE 132        // 128 + 4 pad: float4-aligned rows, conflict-free b64 reads

// ---------------------------------------------------------------------------
// Edge scatter: one 32-lane group per edge; lane L moves float4 chunk L of the
// 128-dim feature row x[src] into agg[dst] with float atomics (resolved in L2,
// since x/agg fit in the 192MB L2). Lane 0 also bumps the in-degree count.
// ---------------------------------------------------------------------------
template <bool WITH_CNT>
__global__ __launch_bounds__(256) void scatter_accum(
    const float* __restrict__ x, const long long* __restrict__ src,
    const long long* __restrict__ dst, float* __restrict__ agg,
    float* __restrict__ cnt, int E)
{
  long long t = (long long)blockIdx.x * blockDim.x + threadIdx.x;
  int e    = (int)(t >> 5);
  int lane = (int)(t & 31);
  if (e >= E) return;
  int s = (int)src[e];
  int d = (int)dst[e];
  const float4 v = ((const float4*)(x + (size_t)s * D_FEAT))[lane];   // coalesced b128
  float* ap = agg + (size_t)d * D_FEAT + lane * 4;
  atomicAdd(ap + 0, v.x);
  atomicAdd(ap + 1, v.y);
  atomicAdd(ap + 2, v.z);
  atomicAdd(ap + 3, v.w);
  if (WITH_CNT && lane == 0) atomicAdd(cnt + d, 1.0f);
}

// ---------------------------------------------------------------------------
// Fused SAGE layer: out = relu?( (agg/max(cnt,1)) @ Wl + x @ Wr + b )
// One block = one 16-row node band; wave w computes the 16x16 output tile at
// columns [16w,16w+16) using V_WMMA_F32_16X16X4_F32 (exact fp32).
//
// Fragment mapping (ISA 7.12.2, 32-bit):
//   A 16x4 : lane(0..15)=row M, half=lane>>4 selects K+{0,2}, vgpr j adds +j
//   B 4x16 : lane(0..15)=col N, half selects K+{0,2}, vgpr j adds +j
//   C/D    : vgpr j -> row j (lanes 0-15) / row 8+j (lanes 16-31), col = lane&15
// ---------------------------------------------------------------------------
template <int COLTILES, bool RELU>
__global__ __launch_bounds__(32 * COLTILES) void sage_layer(
    const float* __restrict__ agg,   // [N,128] summed neighbor features
    const float* __restrict__ cnt,   // [N]     in-degree (float)
    const float* __restrict__ xin,   // [N,128] root features
    const float* __restrict__ Wl,    // [128, DOUT] row-major
    const float* __restrict__ Wr,    // [128, DOUT] row-major
    const float* __restrict__ bias,  // [DOUT]
    float* __restrict__ out)         // [N, DOUT]
{
  constexpr int DOUT = COLTILES * 16;
  __shared__ float lds_mean[16 * LDS_STRIDE];
  __shared__ float lds_x[16 * LDS_STRIDE];
  __shared__ float lds_scale[16];

  const int tid  = threadIdx.x;
  const int row0 = blockIdx.x * 16;

  if (tid < 16) {
    float c = cnt[row0 + tid];
    lds_scale[tid] = 1.0f / fmaxf(c, 1.0f);
  }
  __syncthreads();

  // Stage 16x128 A-tiles (mean and x) into LDS as float4 (512 vec4 per matrix).
  for (int idx = tid; idx < 16 * 32; idx += 32 * COLTILES) {
    int r  = idx >> 5;
    int c4 = idx & 31;
    float  sc = lds_scale[r];
    float4 va = ((const float4*)(agg + (size_t)(row0 + r) * D_FEAT))[c4];
    va.x *= sc; va.y *= sc; va.z *= sc; va.w *= sc;
    *(float4*)&lds_mean[r * LDS_STRIDE + c4 * 4] = va;
    float4 vx = ((const float4*)(xin + (size_t)(row0 + r) * D_FEAT))[c4];
    *(float4*)&lds_x[r * LDS_STRIDE + c4 * 4] = vx;
  }
  __syncthreads();

  const int wave = tid >> 5;
  const int lane = tid & 31;
  const int half = lane >> 4;   // selects K pair {0,1} vs {2,3}
  const int l16  = lane & 15;
  const int col  = wave * 16 + l16;

  v8f acc;
  const float bv = bias[col];
#pragma unroll
  for (int j = 0; j < 8; ++j) acc[j] = bv;

#pragma unroll 4
  for (int k0 = 0; k0 < D_FEAT; k0 += 4) {
    const int ka = k0 + 2 * half;

    v2f a1 = *(const v2f*)&lds_mean[l16 * LDS_STRIDE + ka];   // ds_load_b64
    v2f b1;
    b1.x = Wl[(size_t)ka * DOUT + col];
    b1.y = Wl[(size_t)(ka + 1) * DOUT + col];
    acc = __builtin_amdgcn_wmma_f32_16x16x4_f32(
        false, a1, false, b1, (short)0, acc, false, false);

    v2f a2 = *(const v2f*)&lds_x[l16 * LDS_STRIDE + ka];
    v2f b2;
    b2.x = Wr[(size_t)ka * DOUT + col];
    b2.y = Wr[(size_t)(ka + 1) * DOUT + col];
    acc = __builtin_amdgcn_wmma_f32_16x16x4_f32(
        false, a2, false, b2, (short)0, acc, false, false);
  }

#pragma unroll
  for (int j = 0; j < 8; ++j) {
    float v = acc[j];
    if (RELU) v = fmaxf(v, 0.0f);
    const int m = j + 8 * half;               // D row within tile
    out[(size_t)(row0 + m) * DOUT + col] = v;
  }
}

// ---------------------------------------------------------------------------
// Host launcher (graph-capture safe: only async stream ops).
// Inputs: x, edge_index(int64 [2,E]), W1_l, b1, W1_r, W2_l, b2, W2_r
// ---------------------------------------------------------------------------
extern "C" void kernel_launch(void* const* d_in, const int* in_sizes, int n_in,
                              void* d_out, int out_size, void* d_ws, size_t ws_size,
                              hipStream_t stream)
{
  (void)n_in; (void)out_size; (void)ws_size;
  const float*      x   = (const float*)d_in[0];
  const long long*  ei  = (const long long*)d_in[1];
  const float*      W1l = (const float*)d_in[2];
  const float*      b1  = (const float*)d_in[3];
  const float*      W1r = (const float*)d_in[4];
  const float*      W2l = (const float*)d_in[5];
  const float*      b2  = (const float*)d_in[6];
  const float*      W2r = (const float*)d_in[7];
  float*            out = (float*)d_out;

  const int N = in_sizes[0] / D_FEAT;   // 50000
  const int E = in_sizes[1] / 2;        // 800000
  const long long* src = ei;
  const long long* dst = ei + E;

  // Workspace carve-up (256B-aligned slots): agg[N,128], cnt[N], h[N,128]
  char*  ws  = (char*)d_ws;
  size_t off = 0;
  float* agg = (float*)(ws + off);
  off += ((size_t)N * D_FEAT * sizeof(float) + 255) & ~(size_t)255;
  float* cnt = (float*)(ws + off);
  off += ((size_t)N * sizeof(float) + 255) & ~(size_t)255;
  float* h   = (float*)(ws + off);

  const long long scatterThreads = (long long)E * 32;
  const int scatterBlocks = (int)((scatterThreads + 255) / 256);
  const int bands = N / 16;            // N = 50000 is an exact multiple of 16

  // ---- Layer 1 ----
  hipMemsetAsync(agg, 0, (size_t)N * D_FEAT * sizeof(float), stream);
  hipMemsetAsync(cnt, 0, (size_t)N * sizeof(float), stream);
  scatter_accum<true><<<scatterBlocks, 256, 0, stream>>>(x, src, dst, agg, cnt, E);
  sage_layer<8, true><<<bands, 256, 0, stream>>>(agg, cnt, x, W1l, W1r, b1, h);

  // ---- Layer 2 (dropout is identity in eval; cnt unchanged) ----
  hipMemsetAsync(agg, 0, (size_t)N * D_FEAT * sizeof(float), stream);
  scatter_accum<false><<<scatterBlocks, 256, 0, stream>>>(h, src, dst, agg, cnt, E);
  sage_layer<4, false><<<bands, 128, 0, stream>>>(agg, cnt, h, W2l, W2r, b2, out);
}